// InvertedDotProductAttentionKeyPerQuery_22428319220321
// MI455X (gfx1250) — compile-verified
//
#include <hip/hip_runtime.h>

#define B_  32
#define Qn  16
#define Kn  4096
#define Dn  128
#define Vv  128

typedef __attribute__((ext_vector_type(2))) float v2f;
typedef __attribute__((ext_vector_type(4))) float v4f;
typedef __attribute__((ext_vector_type(8))) float v8f;

// ---------------------------------------------------------------------------
// Phase 1: logits + softmax over the Q axis.
// One thread owns one (b, k): it computes all 16 q-logits (so the softmax
// over Q is entirely thread-local), then writes 16 attn values.
// key (1 GiB, single-use) is streamed with nontemporal float4 loads.
// ---------------------------------------------------------------------------
__global__ __launch_bounds__(256) void invattn_softmax_kernel(
    const float* __restrict__ query,   // [B, Q, D]
    const float* __restrict__ key,     // [B, Q, K, D]
    float* __restrict__ attn)          // [B, Q, K]
{
    __shared__ float qs[Qn * Dn];      // 8 KB: scaled query block for this b

    const int b = blockIdx.y;
    const int k = blockIdx.x * 256 + threadIdx.x;

    // Stage query[b] * 1/sqrt(D) into LDS cooperatively.
    const float scale = 0.08838834764831845f;   // 1/sqrt(128)
    for (int i = threadIdx.x; i < Qn * Dn; i += 256)
        qs[i] = query[(size_t)b * Qn * Dn + i] * scale;
    __syncthreads();

    float logits[Qn];

    #pragma unroll
    for (int q = 0; q < Qn; ++q) {
        const float* kp = key + (((size_t)(b * Qn + q)) * Kn + k) * Dn;
        float ax = 0.f, ay = 0.f, az = 0.f, aw = 0.f;
        #pragma unroll 8
        for (int d = 0; d < Dn; d += 4) {
            v4f kv = __builtin_nontemporal_load((const v4f*)(kp + d));
            v4f qv = *(const v4f*)(qs + q * Dn + d);
            ax += kv.x * qv.x;
            ay += kv.y * qv.y;
            az += kv.z * qv.z;
            aw += kv.w * qv.w;
        }
        logits[q] = (ax + ay) + (az + aw);
    }

    // Softmax over the 16 queries (TEMPERATURE == 1), all in-registers.
    float m = logits[0];
    #pragma unroll
    for (int q = 1; q < Qn; ++q) m = fmaxf(m, logits[q]);
    float s = 0.f;
    #pragma unroll
    for (int q = 0; q < Qn; ++q) {
        logits[q] = __expf(logits[q] - m);
        s += logits[q];
    }
    const float inv = 1.0f / s;
    #pragma unroll
    for (int q = 0; q < Qn; ++q)
        attn[((size_t)(b * Qn + q)) * Kn + k] = logits[q] * inv;
}

// ---------------------------------------------------------------------------
// Phase 2: out[b] (16x128) = attn[b] (16x4096) @ value[b] (4096x128)
// via V_WMMA_F32_16X16X4_F32. One wave per (b, 16-wide N tile);
// 1024 accumulating WMMAs over K.
//
// A 16x4 f32 layout : lane l -> M = l&15 ; lanes 0-15 hold K={0,1},
//                     lanes 16-31 hold K={2,3} (VGPR0/VGPR1).
// B 4x16  f32 layout: lane l -> N = l&15 ; lanes 0-15 hold K={0,1},
//                     lanes 16-31 hold K={2,3}.
// C/D 16x16 f32     : lane l, VGPR r -> M = r + 8*(l>>4), N = l&15.
// ---------------------------------------------------------------------------
__global__ __launch_bounds__(256) void invattn_av_wmma_kernel(
    const float* __restrict__ attn,    // [B, Q, K]
    const float* __restrict__ value,   // [B, K, V]
    float* __restrict__ out)           // [B, Q, V]
{
    const int b     = blockIdx.x;
    const int ntile = threadIdx.x >> 5;        // 0..7 -> N = ntile*16 ..
    const int lane  = threadIdx.x & 31;
    const int l16   = lane & 15;
    const int half  = lane >> 4;               // 0 or 1
    const int kk    = half * 2;                // K sub-offset {0,2}

    const float* ap = attn  + ((size_t)(b * Qn + l16)) * Kn + kk;
    const float* vp = value + ((size_t)b * Kn + kk) * Vv + ntile * 16 + l16;

    v8f c = {0.f, 0.f, 0.f, 0.f, 0.f, 0.f, 0.f, 0.f};

    #pragma unroll 4
    for (int k = 0; k < Kn; k += 4) {
        // A fragment: two consecutive attn values along K (8B aligned).
        v2f a = *(const v2f*)(ap + k);
        // B fragment: two value rows (K and K+1), same column.
        v2f bm;
        bm.x = vp[(size_t)k * Vv];
        bm.y = vp[(size_t)(k + 1) * Vv];
        c = __builtin_amdgcn_wmma_f32_16x16x4_f32(
                /*neg_a=*/false, a, /*neg_b=*/false, bm,
                /*c_mod=*/(short)0, c, /*reuse_a=*/false, /*reuse_b=*/false);
    }

    const int n     = ntile * 16 + l16;
    const int mbase = half * 8;
    float* op = out + (size_t)b * Qn * Vv;
    #pragma unroll
    for (int r = 0; r < 8; ++r)
        op[(size_t)(mbase + r) * Vv + n] = c[r];
}

// ---------------------------------------------------------------------------
extern "C" void kernel_launch(void* const* d_in, const int* in_sizes, int n_in,
                              void* d_out, int out_size, void* d_ws, size_t ws_size,
                              hipStream_t stream) {
    const float* query = (const float*)d_in[0];   // [B,Q,D]
    const float* key   = (const float*)d_in[1];   // [B,Q,K,D]
    const float* value = (const float*)d_in[2];   // [B,K,V]
    // d_in[3] = train flag, unused.

    float* out  = (float*)d_out;                      // [B,Q,V] first
    float* attn = out + (size_t)B_ * Qn * Vv;         // [B,Q,K] second

    dim3 g1(Kn / 256, B_);
    invattn_softmax_kernel<<<g1, 256, 0, stream>>>(query, key, attn);

    invattn_av_wmma_kernel<<<B_, 256, 0, stream>>>(attn, value, out);
}